// Model_76562087018929
// MI455X (gfx1250) — compile-verified
//
#include <hip/hip_runtime.h>
#include <hip/hip_bf16.h>

typedef __attribute__((ext_vector_type(2))) float v2f;
typedef __attribute__((ext_vector_type(8))) float v8f;

#define HID  128
#define LIN  256
#define NCLS 10
#define NG   512

// ---------------- utility ----------------
__global__ void fill_kernel(float* __restrict__ p, float v, size_t n) {
  size_t i  = (size_t)blockIdx.x * blockDim.x + threadIdx.x;
  size_t st = (size_t)gridDim.x * blockDim.x;
  for (; i < n; i += st) p[i] = v;
}

// deg[c] += 1 for each edge (self-loop handled by init deg=1)
__global__ void deg_kernel(const int* __restrict__ col, float* __restrict__ deg, int E) {
  int e = blockIdx.x * blockDim.x + threadIdx.x;
  if (e < E) atomicAdd(&deg[col[e]], 1.0f);
}

__global__ void rsqrt_kernel(float* __restrict__ d, int n) {
  int i = blockIdx.x * blockDim.x + threadIdx.x;
  if (i < n) d[i] = rsqrtf(d[i]);
}

// ---------------- WMMA f32 GEMM: out[nrows x NCOLS] = in[nrows x 128] @ W[128 x NCOLS]
// Block = 8 waves, all on one 64-column group (blockIdx.y); W slice staged in LDS.
// LDS layout pairs consecutive K rows so a B fragment is one ds_load_b64:
//   W[kk][col0+c]  ->  Ws[(kk>>1)*128 + c*2 + (kk&1)]
// f32 16x16x4 operand layout (ISA 7.12.2):
//   A 16x4 : lane m=lane&15 holds row m; lanes 0-15 K={k,k+1}, lanes 16-31 K={k+2,k+3}
//   B 4x16 : lane n=lane&15 holds col n; same K split across lane halves
//   C 16x16: reg r -> row r (lanes 0-15) / row r+8 (lanes 16-31), col = lane&15
template<int NCOLS>
__global__ __launch_bounds__(256) void gemm_wmma_f32(const float* __restrict__ in,
                                                     const float* __restrict__ W,
                                                     float* __restrict__ out, int nrows) {
  constexpr int K = 128;
  __shared__ float Ws[K * 64];                 // 32 KB weight slice
  const int col0 = blockIdx.y * 64;
  const int tid  = threadIdx.x;
#pragma unroll
  for (int i = 0; i < 8; ++i) {                // 256 thr * 8 float4 = 128x64 floats
    int f  = tid + i * 256;                    // float4 id, 0..2047
    int kk = f >> 4;                           // K row 0..127
    int c4 = (f & 15) * 4;                     // col within slice
    const float4 v = *(const float4*)(W + (size_t)kk * NCOLS + col0 + c4);
    float* d = Ws + (kk >> 1) * 128 + (kk & 1) + c4 * 2;
    d[0] = v.x; d[2] = v.y; d[4] = v.z; d[6] = v.w;
  }
  __syncthreads();

  const int wave = tid >> 5;
  const int lane = tid & 31;
  const int rowTile = blockIdx.x * 8 + wave;
  if (rowTile * 16 >= nrows) return;
  const int row0 = rowTile * 16;
  const int m    = lane & 15;                  // A row / B col within tile
  const int koff = (lane >> 4) << 1;           // lane-half K offset {0,2}

  v8f acc[4] = {};
  const float* arow  = in + (size_t)(row0 + m) * K + koff;
  const float* bbase = Ws + m * 2 + (koff >> 1) * 128;
#pragma unroll
  for (int k = 0; k < K; k += 4) {
    const float2 av = *(const float2*)(arow + k);
    v2f a; a.x = av.x; a.y = av.y;
#pragma unroll
    for (int t = 0; t < 4; ++t) {
      const float2 bv = *(const float2*)(bbase + (k >> 1) * 128 + t * 32);
      v2f b; b.x = bv.x; b.y = bv.y;
      acc[t] = __builtin_amdgcn_wmma_f32_16x16x4_f32(false, a, false, b,
                                                     (short)0, acc[t], false, false);
    }
  }
  const int rhalf = (lane >> 4) << 3;          // 0 or 8
#pragma unroll
  for (int t = 0; t < 4; ++t) {
    float* orow = out + (size_t)(row0 + rhalf) * NCOLS + col0 + t * 16 + m;
#pragma unroll
    for (int rg = 0; rg < 8; ++rg)
      orow[(size_t)rg * NCOLS] = acc[t][rg];
  }
}

// ---------------- edge gather/scale/scatter-add: one wave per edge ----------------
__global__ void edge_agg(const int* __restrict__ row, const int* __restrict__ col,
                         const float* __restrict__ dinv, const float* __restrict__ h,
                         float* __restrict__ out, int E) {
  unsigned gtid = blockIdx.x * blockDim.x + threadIdx.x;
  int wave = __builtin_amdgcn_readfirstlane((int)(gtid >> 5));   // wave-uniform -> scalar loads
  int lane = threadIdx.x & 31;
  if (wave >= E) return;
  int r = row[wave];
  int c = col[wave];
  float nrm = dinv[r] * dinv[c];
  float4 v = ((const float4*)(h + (size_t)r * HID))[lane];       // b128 gather, L2-resident
  float* op = out + (size_t)c * HID + lane * 4;
  atomicAdd(op + 0, v.x * nrm);
  atomicAdd(op + 1, v.y * nrm);
  atomicAdd(op + 2, v.z * nrm);
  atomicAdd(op + 3, v.w * nrm);
}

// self-loop contribution + bias + ReLU (after all edge atomics in stream order)
__global__ void selfloop_bias_relu(const float* __restrict__ h, const float* __restrict__ dinv,
                                   const float* __restrict__ bias, float* __restrict__ out,
                                   int nNodes) {
  size_t idx = (size_t)blockIdx.x * blockDim.x + threadIdx.x;
  if (idx >= (size_t)nNodes * HID) return;
  int node = (int)(idx >> 7);
  int c    = (int)(idx & (HID - 1));
  float d = dinv[node];
  float v = out[idx] + h[idx] * d * d + bias[c];
  out[idx] = fmaxf(v, 0.0f);
}

// ---------------- global mean pool ----------------
__global__ void pool_kernel(const float* __restrict__ h, const int* __restrict__ batch,
                            float* __restrict__ pooled, float* __restrict__ counts, int nNodes) {
  unsigned gtid = blockIdx.x * blockDim.x + threadIdx.x;
  int wave = __builtin_amdgcn_readfirstlane((int)(gtid >> 5));
  int lane = threadIdx.x & 31;
  if (wave >= nNodes) return;
  int g = batch[wave];
  float4 v = ((const float4*)(h + (size_t)wave * HID))[lane];
  float* pp = pooled + (size_t)g * HID + lane * 4;
  atomicAdd(pp + 0, v.x);
  atomicAdd(pp + 1, v.y);
  atomicAdd(pp + 2, v.z);
  atomicAdd(pp + 3, v.w);
  if (lane == 0) atomicAdd(&counts[g], 1.0f);
}

__global__ void pool_div(float* __restrict__ pooled, const float* __restrict__ counts, int nG) {
  int idx = blockIdx.x * blockDim.x + threadIdx.x;
  if (idx >= nG * HID) return;
  pooled[idx] /= fmaxf(counts[idx >> 7], 1.0f);
}

__global__ void bias_relu(float* __restrict__ z, const float* __restrict__ b, int total, int c) {
  int i = blockIdx.x * blockDim.x + threadIdx.x;
  if (i < total) z[i] = fmaxf(z[i] + b[i & (c - 1)], 0.0f);
}

// ---------------- lin2 + log_softmax: one thread per graph ----------------
__global__ void head_kernel(const float* __restrict__ z, const float* __restrict__ W,
                            const float* __restrict__ b, float* __restrict__ out, int nG) {
  int g = blockIdx.x * blockDim.x + threadIdx.x;
  if (g >= nG) return;
  float logits[NCLS];
#pragma unroll
  for (int c = 0; c < NCLS; ++c) logits[c] = b[c];
  const float* zp = z + (size_t)g * LIN;
  for (int k = 0; k < LIN; ++k) {
    float zv = zp[k];
    const float* wr = W + (size_t)k * NCLS;
#pragma unroll
    for (int c = 0; c < NCLS; ++c) logits[c] = fmaf(zv, wr[c], logits[c]);
  }
  float mx = logits[0];
#pragma unroll
  for (int c = 1; c < NCLS; ++c) mx = fmaxf(mx, logits[c]);
  float s = 0.0f;
#pragma unroll
  for (int c = 0; c < NCLS; ++c) s += __expf(logits[c] - mx);
  float lse = mx + __logf(s);
#pragma unroll
  for (int c = 0; c < NCLS; ++c) out[(size_t)g * NCLS + c] = logits[c] - lse;
}

__global__ void tail_zero(float* __restrict__ out, int start, int total) {
  int i = start + blockIdx.x * blockDim.x + threadIdx.x;
  if (i < total) out[i] = 0.0f;
}

// ---------------- launch ----------------
extern "C" void kernel_launch(void* const* d_in, const int* in_sizes, int n_in,
                              void* d_out, int out_size, void* d_ws, size_t ws_size,
                              hipStream_t stream) {
  const float* x      = (const float*)d_in[0];
  const int*   eidx   = (const int*)d_in[1];
  const int*   batch  = (const int*)d_in[2];
  const float* W1     = (const float*)d_in[3];
  const float* b1     = (const float*)d_in[4];
  const float* W2     = (const float*)d_in[5];
  const float* b2     = (const float*)d_in[6];
  const float* W3     = (const float*)d_in[7];
  const float* b3     = (const float*)d_in[8];
  const float* lin1_w = (const float*)d_in[9];
  const float* lin1_b = (const float*)d_in[10];
  const float* lin2_w = (const float*)d_in[11];
  const float* lin2_b = (const float*)d_in[12];
  float* out = (float*)d_out;

  const int n = in_sizes[0] / HID;   // 50000 (multiple of 16)
  const int E = in_sizes[1] / 2;     // 1600000
  const int* row = eidx;
  const int* col = eidx + E;

  float* ws = (float*)d_ws;
  float* deg    = ws; ws += n;
  float* bufA   = ws; ws += (size_t)n * HID;   // GEMM result h = cur @ W
  float* bufB   = ws; ws += (size_t)n * HID;   // aggregated layer output
  float* pooled = ws; ws += (size_t)NG * HID;
  float* counts = ws; ws += NG;
  float* z1     = ws; ws += (size_t)NG * LIN;

  // degrees with self-loops, then deg^{-1/2}
  fill_kernel<<<512, 256, 0, stream>>>(deg, 1.0f, (size_t)n);
  deg_kernel<<<(E + 255) / 256, 256, 0, stream>>>(col, deg, E);
  rsqrt_kernel<<<(n + 255) / 256, 256, 0, stream>>>(deg, n);

  const float* layerW[3] = {W1, W2, W3};
  const float* layerB[3] = {b1, b2, b3};
  const float* cur = x;
  const int rowTiles   = n / 16;
  const dim3 gemmGrid((rowTiles + 7) / 8, HID / 64);
  const int aggBlocks  = (E + 7) / 8;               // 8 waves/block, wave per edge
  const int elemBlocks = (int)(((size_t)n * HID + 255) / 256);
  for (int l = 0; l < 3; ++l) {
    gemm_wmma_f32<HID><<<gemmGrid, 256, 0, stream>>>(cur, layerW[l], bufA, n);
    fill_kernel<<<2048, 256, 0, stream>>>(bufB, 0.0f, (size_t)n * HID);
    edge_agg<<<aggBlocks, 256, 0, stream>>>(row, col, deg, bufA, bufB, E);
    selfloop_bias_relu<<<elemBlocks, 256, 0, stream>>>(bufA, deg, layerB[l], bufB, n);
    cur = bufB;
  }

  // global mean pool
  fill_kernel<<<64, 256, 0, stream>>>(pooled, 0.0f, (size_t)NG * HID);
  fill_kernel<<<2, 256, 0, stream>>>(counts, 0.0f, (size_t)NG);
  pool_kernel<<<(n + 7) / 8, 256, 0, stream>>>(bufB, batch, pooled, counts, n);
  pool_div<<<(NG * HID + 255) / 256, 256, 0, stream>>>(pooled, counts, NG);

  // lin1 (WMMA) + bias/ReLU
  const dim3 linGrid((NG / 16 + 7) / 8, LIN / 64);
  gemm_wmma_f32<LIN><<<linGrid, 256, 0, stream>>>(pooled, lin1_w, z1, NG);
  bias_relu<<<(NG * LIN + 255) / 256, 256, 0, stream>>>(z1, lin1_b, NG * LIN, LIN);

  // lin2 + log_softmax
  head_kernel<<<(NG + 255) / 256, 256, 0, stream>>>(z1, lin2_w, lin2_b, out, NG);

  // second tuple element (scalar 0)
  int tail = out_size - NG * NCLS;
  if (tail > 0) tail_zero<<<(tail + 255) / 256, 256, 0, stream>>>(out, NG * NCLS, out_size);
}